// LaGraphNet_28269474742476
// MI455X (gfx1250) — compile-verified
//
#include <hip/hip_runtime.h>

#define N_NODES 200000
#define N_EDGES 6400000
#define N_GRAPHS 512
#define DIM_IN 128
#define HID 32
#define BN_EPS 1e-5f

typedef float v2f __attribute__((ext_vector_type(2)));
typedef float v8f __attribute__((ext_vector_type(8)));

// ---------------------------------------------------------------- utilities
__global__ void fill_zero_kernel(float* __restrict__ p, long long n) {
  long long i = (long long)blockIdx.x * blockDim.x + threadIdx.x;
  if (i < n) p[i] = 0.0f;
}

// ------------------------------------------------------- edge scatter (GIN agg)
// agg[dst] += x[src], vectorized float4 per thread. L2-resident random access.
__global__ void edge_scatter_kernel(const float* __restrict__ x,
                                    const int* __restrict__ src,
                                    const int* __restrict__ dst,
                                    float* __restrict__ agg,
                                    int dim, int vshift, long long total) {
  long long t = (long long)blockIdx.x * blockDim.x + threadIdx.x;
  if (t >= total) return;
  int e = (int)(t >> vshift);
  int j = ((int)t & ((1 << vshift) - 1)) << 2;
  int s = src[e], d = dst[e];
  const float4 v = *(const float4*)(x + (long long)s * dim + j);
  float* ap = agg + (long long)d * dim + j;
  atomicAdd(ap + 0, v.x);
  atomicAdd(ap + 1, v.y);
  atomicAdd(ap + 2, v.z);
  atomicAdd(ap + 3, v.w);
}

// --------------------------------------------------- fused GIN MLP via WMMA f32
// One wave per 16-node tile. pre = x + agg staged in LDS.
// GEMM1: (16 x K) @ (K x 32) + b1, relu -> LDS
// GEMM2: (16 x 32) @ (32 x Nout) + b2, relu -> out; optional BN-stat reduction.
__global__ __launch_bounds__(128)
void gin_mlp_kernel(const float* __restrict__ xin,
                    const float* __restrict__ agg,
                    int K,
                    const float* __restrict__ w1p, const float* __restrict__ b1p,
                    const float* __restrict__ w2p, const float* __restrict__ b2p,
                    int Nout,
                    float* __restrict__ outp,
                    float* __restrict__ stats) {
  __shared__ float sX[4][16 * 132];   // per-wave input tile (padded stride K+4)
  __shared__ float sY[4][16 * 36];    // per-wave hidden tile (stride 36)
  __shared__ float ssum[HID];
  __shared__ float ssq[HID];

  const int tid  = threadIdx.x;
  const int w    = tid >> 5;
  const int lane = tid & 31;
  const int tile = blockIdx.x * 4 + w;
  const int node0 = tile * 16;
  float* X = &sX[w][0];
  float* Y = &sY[w][0];
  const int SW = K + 4;

  const bool do_stats = (stats != nullptr);
  if (do_stats) {
    if (tid < HID) { ssum[tid] = 0.f; ssq[tid] = 0.f; }
    __syncthreads();
  }

  // stage pre-activation tile: x + agg (coalesced along feature dim)
  for (int i = lane; i < 16 * K; i += 32) {
    int r = i / K, c = i - r * K;
    long long g = (long long)(node0 + r) * K + c;
    X[r * SW + c] = xin[g] + agg[g];
  }

  const int m     = lane & 15;        // row (A) / col (B,C,D) within 16
  const int koff  = (lane >> 4) * 2;  // upper half-wave handles K+2,K+3
  const int rbase = (lane >> 4) * 8;  // C/D rows: r + 8*(lane>=16)

  // ---- GEMM1 ----
  v8f c0 = {0.f,0.f,0.f,0.f,0.f,0.f,0.f,0.f};
  v8f c1 = {0.f,0.f,0.f,0.f,0.f,0.f,0.f,0.f};
  for (int k = 0; k < K; k += 4) {
    v2f a; a.x = X[m * SW + k + koff]; a.y = X[m * SW + k + koff + 1];
    const float* wr = w1p + (k + koff) * HID;
    v2f b0; b0.x = wr[m];      b0.y = wr[HID + m];
    v2f b1; b1.x = wr[16 + m]; b1.y = wr[HID + 16 + m];
    c0 = __builtin_amdgcn_wmma_f32_16x16x4_f32(false, a, false, b0, (short)0, c0, false, false);
    c1 = __builtin_amdgcn_wmma_f32_16x16x4_f32(false, a, false, b1, (short)0, c1, false, false);
  }
  for (int r = 0; r < 8; ++r) {
    int rr = rbase + r;
    float y0 = fmaxf(c0[r] + b1p[m], 0.f);
    float y1 = fmaxf(c1[r] + b1p[16 + m], 0.f);
    Y[rr * 36 + m]      = y0;
    Y[rr * 36 + 16 + m] = y1;
  }

  // ---- GEMM2 (loop over 16-wide output tiles; Nout = 32 or 128) ----
  const int NT = Nout >> 4;
  for (int nt = 0; nt < NT; ++nt) {
    v8f c = {0.f,0.f,0.f,0.f,0.f,0.f,0.f,0.f};
    for (int k = 0; k < HID; k += 4) {
      v2f a; a.x = Y[m * 36 + k + koff]; a.y = Y[m * 36 + k + koff + 1];
      const float* wr = w2p + (k + koff) * Nout + nt * 16;
      v2f b; b.x = wr[m]; b.y = wr[Nout + m];
      c = __builtin_amdgcn_wmma_f32_16x16x4_f32(false, a, false, b, (short)0, c, false, false);
    }
    int col = nt * 16 + m;
    float bb = b2p[col];
    float s = 0.f, sq = 0.f;
    for (int r = 0; r < 8; ++r) {
      int rr = rbase + r;
      float v = fmaxf(c[r] + bb, 0.f);   // outer ReLU (applied in every layer)
      outp[(long long)(node0 + rr) * Nout + col] = v;
      s += v; sq += v * v;
    }
    if (do_stats) { atomicAdd(&ssum[col], s); atomicAdd(&ssq[col], sq); }
  }

  if (do_stats) {
    __syncthreads();
    if (tid < HID) {
      atomicAdd(&stats[tid],       ssum[tid]);
      atomicAdd(&stats[HID + tid], ssq[tid]);
    }
  }
}

// -------------------------------------------------- BatchNorm + global-add-pool
__global__ void bn_pool_kernel(const float* __restrict__ h,
                               const float* __restrict__ stats,
                               const float* __restrict__ gamma,
                               const float* __restrict__ beta,
                               float* __restrict__ outp,
                               const int* __restrict__ batch,
                               float* __restrict__ grep, int layer) {
  int idx = blockIdx.x * blockDim.x + threadIdx.x;
  if (idx >= N_NODES * HID) return;
  int f = idx & (HID - 1);
  int node = idx >> 5;
  const float invN = 1.0f / (float)N_NODES;
  float mean = stats[f] * invN;
  float var  = stats[HID + f] * invN - mean * mean;   // biased variance
  float sc   = gamma[f] * rsqrtf(var + BN_EPS);
  float v = (h[idx] - mean) * sc + beta[f];
  outp[idx] = v;
  if (grep) atomicAdd(&grep[(long long)batch[node] * (5 * HID) + layer * HID + f], v);
}

// ------------------------------------------------------------------ host side
static inline void fillz(float* p, long long n, hipStream_t s) {
  int blocks = (int)((n + 255) / 256);
  fill_zero_kernel<<<blocks, 256, 0, s>>>(p, n);
}

static inline void scatter(const float* x, const int* src, const int* dst,
                           float* agg, int dim, hipStream_t s) {
  int vshift = (dim == DIM_IN) ? 5 : 3;   // dim/4 = 32 or 8
  long long total = (long long)N_EDGES << vshift;
  int blocks = (int)((total + 255) / 256);
  edge_scatter_kernel<<<blocks, 256, 0, s>>>(x, src, dst, agg, dim, vshift, total);
}

extern "C" void kernel_launch(void* const* d_in, const int* in_sizes, int n_in,
                              void* d_out, int out_size, void* d_ws, size_t ws_size,
                              hipStream_t stream) {
  (void)in_sizes; (void)n_in; (void)out_size; (void)ws_size;
  const float* x          = (const float*)d_in[0];
  const int*   ei         = (const int*)d_in[1];
  const int*   batch      = (const int*)d_in[2];
  const float* e0_w1      = (const float*)d_in[3];
  const float* e0_b1      = (const float*)d_in[4];
  const float* e0_w2      = (const float*)d_in[5];
  const float* e0_b2      = (const float*)d_in[6];
  const float* enc_w1     = (const float*)d_in[7];
  const float* enc_b1     = (const float*)d_in[8];
  const float* enc_w2     = (const float*)d_in[9];
  const float* enc_b2     = (const float*)d_in[10];
  const float* enc_gamma  = (const float*)d_in[11];
  const float* enc_beta   = (const float*)d_in[12];
  const float* dec_w1     = (const float*)d_in[13];
  const float* dec_b1     = (const float*)d_in[14];
  const float* dec_w2     = (const float*)d_in[15];
  const float* dec_b2     = (const float*)d_in[16];
  const float* dec_w2l    = (const float*)d_in[17];
  const float* dec_b2l    = (const float*)d_in[18];
  const float* dec_gamma  = (const float*)d_in[19];
  const float* dec_beta   = (const float*)d_in[20];

  const int* src = ei;
  const int* dst = ei + N_EDGES;

  // d_out: encoded [N,32] | decoded [N,128] | global_rep [512,160]
  float* enc_out = (float*)d_out;
  float* dec_out = enc_out + (long long)N_NODES * HID;
  float* grep    = dec_out + (long long)N_NODES * DIM_IN;

  // workspace: agg[N*128] | hA[N*32] | hB[N*32] | stats[64]  (~154 MB)
  float* agg   = (float*)d_ws;
  float* hA    = agg + (long long)N_NODES * DIM_IN;
  float* hB    = hA  + (long long)N_NODES * HID;
  float* stats = hB  + (long long)N_NODES * HID;

  const int mlp_blocks = N_NODES / 64;                 // 4 waves/block, 16 nodes/wave
  const int bn_blocks  = (N_NODES * HID + 255) / 256;

  fillz(grep, (long long)N_GRAPHS * 5 * HID, stream);

  // ---------------- encoder layer 0 (K = 128) ----------------
  fillz(agg, (long long)N_NODES * DIM_IN, stream);
  scatter(x, src, dst, agg, DIM_IN, stream);
  fillz(stats, 2 * HID, stream);
  gin_mlp_kernel<<<mlp_blocks, 128, 0, stream>>>(x, agg, DIM_IN,
      e0_w1, e0_b1, e0_w2, e0_b2, HID, hA, stats);
  bn_pool_kernel<<<bn_blocks, 256, 0, stream>>>(hA, stats,
      enc_gamma, enc_beta, hB, batch, grep, 0);
  const float* cur = hB;

  // ---------------- encoder layers 1..4 ----------------
  for (int i = 0; i < 4; ++i) {
    fillz(agg, (long long)N_NODES * HID, stream);
    scatter(cur, src, dst, agg, HID, stream);
    fillz(stats, 2 * HID, stream);
    gin_mlp_kernel<<<mlp_blocks, 128, 0, stream>>>(cur, agg, HID,
        enc_w1 + i * HID * HID, enc_b1 + i * HID,
        enc_w2 + i * HID * HID, enc_b2 + i * HID, HID, hA, stats);
    float* o = (i == 3) ? enc_out : hB;
    bn_pool_kernel<<<bn_blocks, 256, 0, stream>>>(hA, stats,
        enc_gamma + (i + 1) * HID, enc_beta + (i + 1) * HID, o, batch, grep, i + 1);
    cur = o;
  }

  // ---------------- decoder layers 0..3 (BN, no pool) ----------------
  for (int i = 0; i < 4; ++i) {
    fillz(agg, (long long)N_NODES * HID, stream);
    scatter(cur, src, dst, agg, HID, stream);
    fillz(stats, 2 * HID, stream);
    gin_mlp_kernel<<<mlp_blocks, 128, 0, stream>>>(cur, agg, HID,
        dec_w1 + i * HID * HID, dec_b1 + i * HID,
        dec_w2 + i * HID * HID, dec_b2 + i * HID, HID, hA, stats);
    bn_pool_kernel<<<bn_blocks, 256, 0, stream>>>(hA, stats,
        dec_gamma + i * HID, dec_beta + i * HID, hB, batch, nullptr, 0);
    cur = hB;
  }

  // ---------------- decoder layer 4 (Nout = 128, relu only) ----------------
  fillz(agg, (long long)N_NODES * HID, stream);
  scatter(cur, src, dst, agg, HID, stream);
  gin_mlp_kernel<<<mlp_blocks, 128, 0, stream>>>(cur, agg, HID,
      dec_w1 + 4 * HID * HID, dec_b1 + 4 * HID,
      dec_w2l, dec_b2l, DIM_IN, dec_out, nullptr);
}